// PositionSensitiveScoreMap_87917980549144
// MI455X (gfx1250) — compile-verified
//
#include <hip/hip_runtime.h>
#include <math.h>

// ---------------------------------------------------------------------------
// Problem constants (from reference): B=4, K=7, c1=21, H=W=100
//   planes = K*K*c1 = 1029, integral plane = 101x101
//   rois: 2000 x 4 int32 (ymin, xmin, ymax, xmax)
//   output: 2000 x 21 f32 softmax scores
// ---------------------------------------------------------------------------
#define KK        7
#define C1        21
#define PLANES    (KK * KK * C1)   // 1029
#define HDIM      100
#define WDIM      100
#define IW        101
#define ISZ       (IW * IW)        // 10201 floats per integral plane

typedef __attribute__((ext_vector_type(2))) float v2f;
typedef __attribute__((ext_vector_type(8))) float v8f;

__device__ __forceinline__ float wave_bcast_or_shift(float v, int srcLane) {
    // wave32 lane gather via ds_bpermute (byte index, bits[6:2] used)
    int r = __builtin_amdgcn_ds_bpermute(srcLane << 2, __float_as_int(v));
    return __int_as_float(r);
}

// ---------------------------------------------------------------------------
// Kernel A: row cumsum with top/left zero padding.
// One wave per (plane, output-row). Wave32 inclusive scan over 100 elements
// in 4 chunks of 32 using ds_bpermute lane shuffles.
// I[p, 0, :] = 0 ; I[p, y, 0] = 0 ; I[p, y, x] = sum_{x'<=x-1} in[p, y-1, x']
// ---------------------------------------------------------------------------
__global__ __launch_bounds__(256) void psroi_row_cumsum(
    const float* __restrict__ in,   // last-batch feature maps: PLANES x 100 x 100
    float* __restrict__ I)          // PLANES x 101 x 101
{
    const int njobs = PLANES * IW;                    // 103929 row jobs
    const int wavesPerBlock = blockDim.x >> 5;
    const int jb = blockIdx.x * wavesPerBlock + (threadIdx.x >> 5);
    if (jb >= njobs) return;                          // wave-uniform exit
    const int lane = threadIdx.x & 31;

    const int p = jb / IW;
    const int y = jb - p * IW;
    float* Ip = I + (size_t)p * ISZ;

    if (y == 0) {                                     // top padding row
        for (int x = lane; x < IW; x += 32) Ip[x] = 0.0f;
        return;
    }

    const float* src = in + (size_t)p * (HDIM * WDIM) + (size_t)(y - 1) * WDIM;
    if (lane == 0) Ip[y * IW] = 0.0f;                 // left padding col

    float carry = 0.0f;
    #pragma unroll
    for (int t = 0; t < 4; ++t) {
        const int x = t * 32 + lane;
        float v = (x < WDIM) ? src[x] : 0.0f;
        // wave32 inclusive scan
        #pragma unroll
        for (int off = 1; off < 32; off <<= 1) {
            float nv = wave_bcast_or_shift(v, lane - off);
            v += (lane >= off) ? nv : 0.0f;
        }
        const float total = wave_bcast_or_shift(v, 31);
        if (x < WDIM) Ip[y * IW + (x + 1)] = carry + v;
        carry += total;
    }
}

// ---------------------------------------------------------------------------
// Kernel B: column cumsum in place. One block per plane, thread per column
// (coalesced across a 101-float row).
// ---------------------------------------------------------------------------
__global__ __launch_bounds__(128) void psroi_col_cumsum(float* __restrict__ I)
{
    const int p = blockIdx.x;
    const int x = threadIdx.x;
    if (x >= IW) return;
    float* Ip = I + (size_t)p * ISZ;
    float running = 0.0f;
    for (int y = 0; y < IW; ++y) {
        const int idx = y * IW + x;
        running += Ip[idx];
        Ip[idx] = running;
    }
}

// ---------------------------------------------------------------------------
// Kernel C: per-ROI PS-ROI pooling + bin reduction via WMMA + softmax.
// One wave (32 threads) per ROI. EXEC is all-ones at every WMMA.
//
// WMMA trick: with B = all-ones, V_WMMA_F32_16X16X4_F32 computes
//   D[m,n] = sum_k A[m,k] + C[m,n]
// i.e. 16 channels x 4 bins summed per instruction, chained through C over
// 13 chunks (52 >= 49 bins, tail zero-padded). Two passes cover 21 channels.
//
// A layout (documented): lanes 0-15 hold M=lane, VGPR0=K0, VGPR1=K1;
//                        lanes 16-31 hold M=lane-16, VGPR0=K2, VGPR1=K3.
// D layout (documented): VGPR r, lanes 0-15 -> D[r, lane]; lanes 16-31 ->
//                        D[r+8, lane-16]. So lane 0 VGPR i = row-i sum,
//                        lane 16 VGPR i = row-(i+8) sum.
// ---------------------------------------------------------------------------
__global__ __launch_bounds__(32) void psroi_pool_softmax(
    const float* __restrict__ I,    // PLANES x 101 x 101 integral images
    const int* __restrict__ rois,   // n x 4
    float* __restrict__ out)        // n x 21
{
    __shared__ float ssum[32];
    __shared__ float sexp[32];

    const int n    = blockIdx.x;
    const int lane = threadIdx.x;          // 0..31, single wave
    const int half = lane >> 4;            // A-matrix K-half
    const int m    = lane & 15;            // A-matrix row (channel-in-block)

    const int ymin = rois[n * 4 + 0];
    const int xmin = rois[n * 4 + 1];
    const int ymax = rois[n * 4 + 2];
    const int xmax = rois[n * 4 + 3];
    const int ys   = (ymax - ymin) / KK;   // >= 1
    const int xs   = (xmax - xmin) / KK;   // >= 1

    const v2f onesB = {1.0f, 1.0f};

    #pragma unroll
    for (int pass = 0; pass < 2; ++pass) {
        const int c = pass * 16 + m;
        const bool cvalid = (c < C1);
        v8f acc = {};                      // C starts at 0

        for (int t = 0; t < 13; ++t) {     // 13 chunks x 4 bins = 52 >= 49
            v2f a;
            #pragma unroll
            for (int r = 0; r < 2; ++r) {
                const int b = t * 4 + half * 2 + r;   // bin index (K dim)
                float s = 0.0f;
                if (cvalid && b < KK * KK) {
                    const int j  = b / KK;
                    const int l  = b - j * KK;
                    const int y0 = ymin + j * ys, y1 = y0 + ys;
                    const int x0 = xmin + l * xs, x1 = x0 + xs;
                    const float* base = I + (size_t)(b * C1 + c) * ISZ;
                    s = base[y1 * IW + x1] - base[y0 * IW + x1]
                      - base[y1 * IW + x0] + base[y0 * IW + x0];
                }
                a[r] = s;
            }
            // D = A x ones(4x16) + C  ->  per-row 4-bin sums accumulate in C
            acc = __builtin_amdgcn_wmma_f32_16x16x4_f32(
                false, a, false, onesB, (short)0, acc, false, false);
        }

        // Extract per-channel bin sums from D column 0.
        if (lane == 0) {
            #pragma unroll
            for (int i = 0; i < 8; ++i) {
                const int c2 = pass * 16 + i;
                if (c2 < C1) ssum[c2] = acc[i];
            }
        }
        if (lane == 16) {
            #pragma unroll
            for (int i = 0; i < 8; ++i) {
                const int c2 = pass * 16 + 8 + i;
                if (c2 < C1) ssum[c2] = acc[i];
            }
        }
    }
    __syncthreads();

    // softmax over 21 channels; avg = binSum / (49 * area)
    const float inv = 1.0f / (49.0f * (float)(ys * xs));
    float vmax = -INFINITY;
    for (int c2 = 0; c2 < C1; ++c2) vmax = fmaxf(vmax, ssum[c2] * inv);

    float e = 0.0f;
    if (lane < C1) e = expf(ssum[lane] * inv - vmax);
    sexp[lane] = e;
    __syncthreads();

    float tot = 0.0f;
    for (int c2 = 0; c2 < C1; ++c2) tot += sexp[c2];
    if (lane < C1) out[n * C1 + lane] = e / tot;
}

// ---------------------------------------------------------------------------
// Host-side launcher
// ---------------------------------------------------------------------------
extern "C" void kernel_launch(void* const* d_in, const int* in_sizes, int n_in,
                              void* d_out, int out_size, void* d_ws, size_t ws_size,
                              hipStream_t stream) {
    const float* cls_conv_out = (const float*)d_in[0];
    const int*   rois         = (const int*)d_in[1];
    float*       out          = (float*)d_out;
    float*       I            = (float*)d_ws;          // PLANES*101*101 f32 (~42 MB)

    const int planeElems = PLANES * HDIM * WDIM;       // 10,290,000
    const int batch      = in_sizes[0] / planeElems;   // 4
    const float* lastBatch = cls_conv_out + (size_t)(batch - 1) * planeElems;

    const int nroi = in_sizes[1] / 4;                  // 2000
    (void)n_in; (void)out_size; (void)ws_size;

    // A: row cumsum (one wave per row job)
    {
        const int njobs = PLANES * IW;                 // 103929
        const int wavesPerBlock = 8;                   // 256 threads
        const int blocks = (njobs + wavesPerBlock - 1) / wavesPerBlock;
        psroi_row_cumsum<<<blocks, 256, 0, stream>>>(lastBatch, I);
    }
    // B: column cumsum (one block per plane)
    psroi_col_cumsum<<<PLANES, 128, 0, stream>>>(I);
    // C: per-ROI pooling + WMMA bin reduction + softmax (one wave per ROI)
    psroi_pool_softmax<<<nroi, 32, 0, stream>>>(I, rois, out);
}